// Feature_Engineering_69260642615649
// MI455X (gfx1250) — compile-verified
//
#include <hip/hip_runtime.h>
#include <hip/hip_bf16.h>

// Problem constants (from reference): x is (4, 32, 30000, 2) fp32,
// out is (4, 656, 30000) fp32.
#define NB 4
#define NCH 16
#define NPT 30000
#define OUT_CH 656

// Pairwise-kernel tiling
#define TILE_N 128
#define TILE_COLS 132                     // staged n-values per row (TILE_N + 4, 16B multiple)
#define ROW_BYTES (TILE_COLS * 8)         // 1056 bytes per channel row (float2)
#define CHUNKS_PER_ROW (ROW_BYTES / 16)   // 66 x 16B chunks
#define NUM_ROWS 32                       // 16 x1-channels + 16 x2-channels
#define NUM_CHUNKS (NUM_ROWS * CHUNKS_PER_ROW)  // 2112
#define X_BYTES (4LL * 32LL * 30000LL * 2LL * 4LL)  // 30,720,000 bytes

// ---------------------------------------------------------------------------
// CDNA5 async global->LDS copy (VGLOBAL GLOBAL_LOAD_ASYNC_TO_LDS_B128).
// vdst VGPR = per-lane LDS byte offset, vaddr = per-lane 64-bit global addr.
// Tracked by ASYNCcnt; drained with s_wait_asynccnt before the barrier.
// ---------------------------------------------------------------------------
__device__ __forceinline__ void async_copy16(unsigned lds_off, const void* gaddr) {
  asm volatile("global_load_async_to_lds_b128 %0, %1, off"
               :: "v"(lds_off), "v"(gaddr)
               : "memory");
}

__device__ __forceinline__ void wait_async0() {
  asm volatile("s_wait_asynccnt 0" ::: "memory");
}

// ---------------------------------------------------------------------------
// Kernel: pairwise distance d (out channels 112..367) and dd (368..623).
// Block = (b, 128-n tile). Stage x1/x2 (all 32 channels, 132 n-values each)
// into LDS via the async-copy engine, then 256 (shift,channel) pairs read the
// tile conflict-free (lane-consecutive float2 -> ds b64 reads).
// Output volume dominates: 246 MB of the 315 MB total -> NT streaming stores.
// ---------------------------------------------------------------------------
__global__ void __launch_bounds__(256)
fe_pair_kernel(const float* __restrict__ x, float* __restrict__ out) {
  __shared__ __align__(16) float2 tile[NUM_ROWS * TILE_COLS];

  const int b   = blockIdx.y;
  const int n0  = blockIdx.x * TILE_N;
  const int tid = threadIdx.x;

  // ---- async stage: 2112 x 16B chunks, 256 threads (~8 async ops/thread) ----
  const char* xb = (const char*)x;
  const unsigned lds_base = (unsigned)(unsigned long long)(void*)&tile[0];
  for (int t = tid; t < NUM_CHUNKS; t += 256) {
    const int row = t / CHUNKS_PER_ROW;
    const int col = t - row * CHUNKS_PER_ROW;
    long long goff = ((long long)(b * 32 + row) * NPT + n0) * 8 + (long long)col * 16;
    const long long gmax = X_BYTES - 16;     // clamp tail (garbage never consumed)
    if (goff > gmax) goff = gmax;
    async_copy16(lds_base + (unsigned)(row * ROW_BYTES + col * 16), xb + goff);
  }
  wait_async0();
  __syncthreads();

  // ---- compute: wave lanes sweep n (coalesced 128B NT stores), k-rows sweep pairs
  const int  lane = tid & 31;
  const int  krow = tid >> 5;                // 0..7
  const bool full = (n0 + TILE_N) < NPT;     // interior tile: no n-guards needed

  for (int kk = krow; kk < 256; kk += 8) {
    const int s  = kk >> 4;                  // shift-1 (shift = s+1)
    const int c  = kk & 15;
    const int a1 = (c - s - 1) & 15;         // rolled x1 channel
    const float2* rowA = &tile[a1 * TILE_COLS];
    const float2* rowB = &tile[(16 + c) * TILE_COLS];
    float* pd  = out + ((size_t)b * OUT_CH + 112 + kk) * NPT + n0;
    float* pdd = out + ((size_t)b * OUT_CH + 368 + kk) * NPT + n0;

    if (full) {
      // Fast path: 234/235 tiles. No bounds or N-1 handling.
#pragma unroll
      for (int nc = 0; nc < TILE_N; nc += 32) {
        const int nn = nc + lane;
        const float2 A  = rowA[nn];
        const float2 A1 = rowA[nn + 1];
        const float2 Bv = rowB[nn];
        const float2 B1 = rowB[nn + 1];
        const float ex0 = A.x - Bv.x + 1e-6f;
        const float ey0 = A.y - Bv.y + 1e-6f;
        const float d0  = sqrtf(ex0 * ex0 + ey0 * ey0);
        const float ex1 = A1.x - B1.x + 1e-6f;
        const float ey1 = A1.y - B1.y + 1e-6f;
        const float d1  = sqrtf(ex1 * ex1 + ey1 * ey1);
        __builtin_nontemporal_store(d0,      pd  + nn);
        __builtin_nontemporal_store(d1 - d0, pdd + nn);
      }
    } else {
      // Boundary tile: guard n < N; dd[N-1] = -d[N-1].
      for (int nc = 0; nc < TILE_N; nc += 32) {
        const int nn = nc + lane;
        const int ng = n0 + nn;
        if (ng >= NPT) continue;
        const float2 A  = rowA[nn];
        const float2 A1 = rowA[nn + 1];
        const float2 Bv = rowB[nn];
        const float2 B1 = rowB[nn + 1];
        const float ex0 = A.x - Bv.x + 1e-6f;
        const float ey0 = A.y - Bv.y + 1e-6f;
        const float d0  = sqrtf(ex0 * ex0 + ey0 * ey0);
        const float ex1 = A1.x - B1.x + 1e-6f;
        const float ey1 = A1.y - B1.y + 1e-6f;
        const float d1  = sqrtf(ex1 * ex1 + ey1 * ey1);
        const float dd  = (ng == NPT - 1) ? -d0 : (d1 - d0);
        __builtin_nontemporal_store(d0, pd  + nn);
        __builtin_nontemporal_store(dd, pdd + nn);
      }
    }
  }
}

// ---------------------------------------------------------------------------
// Kernel: per-channel features (out channels 0..111 and 624..655).
// One thread per (b, c, n); overlapping reads hit L0/L2 (input is 30 MB,
// L2 is 192 MB). 69 MB of NT output stores.
// ---------------------------------------------------------------------------
__global__ void __launch_bounds__(256)
fe_chan_kernel(const float* __restrict__ x, float* __restrict__ out) {
  const int n = blockIdx.x * 256 + threadIdx.x;
  if (n >= NPT) return;
  const int bc = blockIdx.y;          // b*16 + c
  const int b  = bc >> 4;
  const int c  = bc & 15;

  const float2* x1 = (const float2*)x + (size_t)(b * 32 + c) * NPT;
  const float2* x2 = (const float2*)x + (size_t)(b * 32 + 16 + c) * NPT;

  const float2 p0 = x1[n];
  const float2 q0 = x2[n];

  // dx at n and at n-1 (tdiff semantics: dx[0] = 0)
  float dx1x = 0.f, dx1y = 0.f, dx2x = 0.f, dx2y = 0.f;
  float px1x = 0.f, px1y = 0.f, px2x = 0.f, px2y = 0.f;
  if (n >= 1) {
    const float2 p1 = x1[n - 1];
    const float2 q1 = x2[n - 1];
    dx1x = p0.x - p1.x; dx1y = p0.y - p1.y;
    dx2x = q0.x - q1.x; dx2y = q0.y - q1.y;
    if (n >= 2) {
      const float2 p2 = x1[n - 2];
      const float2 q2 = x2[n - 2];
      px1x = p1.x - p2.x; px1y = p1.y - p2.y;
      px2x = q1.x - q2.x; px2y = q1.y - q2.y;
    }
  }
  // ddx (tdiff of dx): zero at n==0 automatically (0-0)
  const float ddx1x = dx1x - px1x, ddx1y = dx1y - px1y;
  const float ddx2x = dx2x - px2x, ddx2y = dx2y - px2y;

  const float n1  = sqrtf(dx1x * dx1x + dx1y * dx1y);
  const float n2  = sqrtf(dx2x * dx2x + dx2y * dx2y);
  const float n1p = sqrtf(px1x * px1x + px1y * px1y);
  const float n2p = sqrtf(px2x * px2x + px2y * px2y);

  const float cp1 = dx1x * ddx1y - dx1y * ddx1x;
  const float cp2 = dx2x * ddx2y - dx2y * ddx2x;

  const float adx1 = (dx1x * px1x + dx1y * px1y) / (n1 * n1p + 1e-4f);
  const float adx2 = (dx2x * px2x + dx2y * px2y) / (n2 * n2p + 1e-4f);
  const float dirs = (dx1x * dx2x + dx1y * dx2y) / (n1 * n2 + 1e-6f);

  const float relx = p0.x - q0.x, rely = p0.y - q0.y;
  const float rn   = sqrtf(relx * relx + rely * rely);
  const float lead1 =  (dx1x * relx + dx1y * rely) / (n1 * rn + 1e-6f);
  const float lead2 = -(dx2x * relx + dx2y * rely) / (n2 * rn + 1e-6f);

  float* ob = out + (size_t)b * OUT_CH * NPT + n;
  __builtin_nontemporal_store(n1,    ob + (size_t)(  0 + c) * NPT);
  __builtin_nontemporal_store(n2,    ob + (size_t)( 16 + c) * NPT);
  __builtin_nontemporal_store(cp1,   ob + (size_t)( 32 + c) * NPT);
  __builtin_nontemporal_store(cp2,   ob + (size_t)( 48 + c) * NPT);
  __builtin_nontemporal_store(adx1,  ob + (size_t)( 64 + c) * NPT);
  __builtin_nontemporal_store(adx2,  ob + (size_t)( 80 + c) * NPT);
  __builtin_nontemporal_store(dirs,  ob + (size_t)( 96 + c) * NPT);
  __builtin_nontemporal_store(lead1, ob + (size_t)(624 + c) * NPT);
  __builtin_nontemporal_store(lead2, ob + (size_t)(640 + c) * NPT);
}

// ---------------------------------------------------------------------------
extern "C" void kernel_launch(void* const* d_in, const int* in_sizes, int n_in,
                              void* d_out, int out_size, void* d_ws, size_t ws_size,
                              hipStream_t stream) {
  (void)in_sizes; (void)n_in; (void)out_size; (void)d_ws; (void)ws_size;
  const float* x = (const float*)d_in[0];   // (4, 32, 30000, 2) fp32
  float* out = (float*)d_out;               // (4, 656, 30000) fp32

  dim3 b2(256), g2((NPT + TILE_N - 1) / TILE_N, NB);  // 235 x 4
  fe_pair_kernel<<<g2, b2, 0, stream>>>(x, out);

  dim3 b1(256), g1((NPT + 255) / 256, NB * NCH);      // 118 x 64
  fe_chan_kernel<<<g1, b1, 0, stream>>>(x, out);
}